// GraphSage_67070209295068
// MI455X (gfx1250) — compile-verified
//
#include <hip/hip_runtime.h>

typedef __attribute__((ext_vector_type(2))) float v2f;
typedef __attribute__((ext_vector_type(8))) float v8f;

#define N_NODES 100000
#define N_EDGES 1600000
#define F_IN    128
#define F_HID   128
#define N_CLS   40

// ---------------- degree: deg[dst[e]] += 1 ----------------
__global__ void sage_degree_kernel(const int* __restrict__ dst,
                                   float* __restrict__ deg, int nE) {
    int e = blockIdx.x * blockDim.x + threadIdx.x;
    if (e < nE) atomicAdd(&deg[dst[e]], 1.0f);
}

// -------- scatter-add: agg[dst[e]][0:128] += X[src[e]][0:128] --------
// one wave (32 lanes) per edge, float4 per lane, 4 fp32 atomics per lane.
// agg (51.2 MB) is L2-resident on MI455X (192 MB L2) so atomics stay on-chip.
__global__ __launch_bounds__(256)
void sage_scatter_kernel(const float* __restrict__ X,
                         const int* __restrict__ src,
                         const int* __restrict__ dst,
                         float* __restrict__ agg, int nE) {
    int t    = blockIdx.x * blockDim.x + threadIdx.x;
    int e    = t >> 5;
    int lane = t & 31;
    if (e >= nE) return;
    int s = src[e];
    int d = dst[e];
    float4 v = ((const float4*)(X + (size_t)s * F_IN))[lane];
    float* o = agg + (size_t)d * F_IN + lane * 4;
    atomicAdd(o + 0, v.x);
    atomicAdd(o + 1, v.y);
    atomicAdd(o + 2, v.z);
    atomicAdd(o + 3, v.w);
}

// -------- layer 1: H = relu(X@Wself + (agg/deg)@Wneigh + b) --------
// block = 256 threads = 8 waves; each wave owns one 16x16 output tile,
// 8 waves tile the full 128-wide output panel for 16 rows.
// WMMA: V_WMMA_F32_16X16X4_F32, K stepped by 4, fp32 end to end.
__global__ __launch_bounds__(256)
void sage_gemm1_kernel(const float* __restrict__ X,      // [N,128]
                       const float* __restrict__ AGG,    // [N,128]
                       const float* __restrict__ deg,    // [N]
                       const float* __restrict__ Wself,  // [128,128]
                       const float* __restrict__ Wneigh, // [128,128]
                       const float* __restrict__ bias,   // [128]
                       float* __restrict__ H)            // [N,128]
{
    const int wave = threadIdx.x >> 5;   // 0..7 -> n tile
    const int lane = threadIdx.x & 31;
    const int half = lane >> 4;          // 0 or 1
    const int r    = lane & 15;
    const int m0   = blockIdx.x * 16;    // 100000 = 16*6250 exactly, no guard
    const int n0   = wave * 16;

    const int   arow = m0 + r;           // A-fragment row for this lane
    const float idg  = 1.0f / fmaxf(deg[arow], 1.0f);
    const float* xr  = X   + (size_t)arow * F_IN;
    const float* gr  = AGG + (size_t)arow * F_IN;

    v8f acc = {};
    // self term: X @ Wself
    #pragma unroll 4
    for (int k0 = 0; k0 < F_IN; k0 += 4) {
        const int ka = k0 + 2 * half;                    // lane's K pair
        v2f a; a.x = xr[ka]; a.y = xr[ka + 1];
        v2f b; b.x = Wself[(size_t)ka       * F_HID + n0 + r];
               b.y = Wself[(size_t)(ka + 1) * F_HID + n0 + r];
        acc = __builtin_amdgcn_wmma_f32_16x16x4_f32(false, a, false, b,
                                                    (short)0, acc, false, false);
    }
    // neighbor term: (agg * 1/deg) @ Wneigh, mean fused into A load
    #pragma unroll 4
    for (int k0 = 0; k0 < F_IN; k0 += 4) {
        const int ka = k0 + 2 * half;
        v2f a; a.x = gr[ka] * idg; a.y = gr[ka + 1] * idg;
        v2f b; b.x = Wneigh[(size_t)ka       * F_HID + n0 + r];
               b.y = Wneigh[(size_t)(ka + 1) * F_HID + n0 + r];
        acc = __builtin_amdgcn_wmma_f32_16x16x4_f32(false, a, false, b,
                                                    (short)0, acc, false, false);
    }

    const int   col = n0 + r;
    const float bv  = bias[col];
    #pragma unroll
    for (int j = 0; j < 8; ++j) {
        const int orow = m0 + j + 8 * half;              // C/D layout: M = vgpr + 8*half
        H[(size_t)orow * F_HID + col] = fmaxf(acc[j] + bv, 0.0f);
    }
}

// -------- layer 2: OUT = H@Wself + (agg/deg)@Wneigh + b  (N=40, pad to 48) --------
__global__ __launch_bounds__(96)
void sage_gemm2_kernel(const float* __restrict__ H,      // [N,128]
                       const float* __restrict__ AGG,    // [N,128]
                       const float* __restrict__ deg,    // [N]
                       const float* __restrict__ Wself,  // [128,40]
                       const float* __restrict__ Wneigh, // [128,40]
                       const float* __restrict__ bias,   // [40]
                       float* __restrict__ OUT)          // [N,40]
{
    const int wave = threadIdx.x >> 5;   // 0..2 -> n tile (48 cols padded)
    const int lane = threadIdx.x & 31;
    const int half = lane >> 4;
    const int r    = lane & 15;
    const int m0   = blockIdx.x * 16;
    const int n0   = wave * 16;

    const int   arow  = m0 + r;
    const float idg   = 1.0f / fmaxf(deg[arow], 1.0f);
    const int   col   = n0 + r;
    const bool  colOK = (col < N_CLS);
    const float* xr   = H   + (size_t)arow * F_HID;
    const float* gr   = AGG + (size_t)arow * F_HID;

    v8f acc = {};
    #pragma unroll 4
    for (int k0 = 0; k0 < F_HID; k0 += 4) {
        const int ka = k0 + 2 * half;
        v2f a; a.x = xr[ka]; a.y = xr[ka + 1];
        v2f b; b.x = colOK ? Wself[(size_t)ka       * N_CLS + col] : 0.0f;
               b.y = colOK ? Wself[(size_t)(ka + 1) * N_CLS + col] : 0.0f;
        acc = __builtin_amdgcn_wmma_f32_16x16x4_f32(false, a, false, b,
                                                    (short)0, acc, false, false);
    }
    #pragma unroll 4
    for (int k0 = 0; k0 < F_HID; k0 += 4) {
        const int ka = k0 + 2 * half;
        v2f a; a.x = gr[ka] * idg; a.y = gr[ka + 1] * idg;
        v2f b; b.x = colOK ? Wneigh[(size_t)ka       * N_CLS + col] : 0.0f;
               b.y = colOK ? Wneigh[(size_t)(ka + 1) * N_CLS + col] : 0.0f;
        acc = __builtin_amdgcn_wmma_f32_16x16x4_f32(false, a, false, b,
                                                    (short)0, acc, false, false);
    }

    if (colOK) {
        const float bv = bias[col];
        #pragma unroll
        for (int j = 0; j < 8; ++j) {
            const int orow = m0 + j + 8 * half;
            OUT[(size_t)orow * N_CLS + col] = acc[j] + bv;
        }
    }
}

extern "C" void kernel_launch(void* const* d_in, const int* in_sizes, int n_in,
                              void* d_out, int out_size, void* d_ws, size_t ws_size,
                              hipStream_t stream) {
    const float* X   = (const float*)d_in[0];
    const int*   src = (const int*)  d_in[1];
    const int*   dst = (const int*)  d_in[2];
    const float* W1s = (const float*)d_in[3];
    const float* W1n = (const float*)d_in[4];
    const float* b1  = (const float*)d_in[5];
    const float* W2s = (const float*)d_in[6];
    const float* W2n = (const float*)d_in[7];
    const float* b2  = (const float*)d_in[8];
    float*       out = (float*)d_out;

    // workspace: deg[N] | agg[N*128] | h1[N*128]  (~103 MB)
    char*  ws     = (char*)d_ws;
    size_t degOff = 0;
    size_t aggOff = ((size_t)N_NODES * sizeof(float) + 255) & ~(size_t)255;
    size_t h1Off  = aggOff + (size_t)N_NODES * F_IN * sizeof(float);
    float* deg = (float*)(ws + degOff);
    float* agg = (float*)(ws + aggOff);
    float* h1  = (float*)(ws + h1Off);

    const size_t aggBytes = (size_t)N_NODES * F_IN * sizeof(float);

    // ---- layer 1 ----
    hipMemsetAsync(deg, 0, (size_t)N_NODES * sizeof(float), stream);
    hipMemsetAsync(agg, 0, aggBytes, stream);
    sage_degree_kernel<<<(N_EDGES + 255) / 256, 256, 0, stream>>>(dst, deg, N_EDGES);
    sage_scatter_kernel<<<(N_EDGES * 32) / 256, 256, 0, stream>>>(X, src, dst, agg, N_EDGES);
    sage_gemm1_kernel<<<N_NODES / 16, 256, 0, stream>>>(X, agg, deg, W1s, W1n, b1, h1);

    // ---- layer 2 (reuse agg buffer) ----
    hipMemsetAsync(agg, 0, aggBytes, stream);
    sage_scatter_kernel<<<(N_EDGES * 32) / 256, 256, 0, stream>>>(h1, src, dst, agg, N_EDGES);
    sage_gemm2_kernel<<<N_NODES / 16, 96, 0, stream>>>(h1, agg, deg, W2s, W2n, b2, out);
}